// mGCN_17927193494277
// MI455X (gfx1250) — compile-verified
//
#include <hip/hip_runtime.h>
#include <hip/hip_bf16.h>

typedef __attribute__((ext_vector_type(2))) float v2f;
typedef __attribute__((ext_vector_type(8))) float v8f;

#define TILE_K_MAX 384

__device__ __forceinline__ float eluf(float x) {
    return x > 0.f ? x : expm1f(x);
}

// ---------------------------------------------------------------------------
// C[M x Ncol] = A[M x K] @ B[Ncol x K]^T  (+ bias[col], optional ELU)
// A may be "view-split": element (n,k) at A + (k/vW)*vS + n*lda + (k%vW)
// One 16x16 output tile per wave via V_WMMA_F32_16X16X4_F32; A tile in LDS.
// Block = 128 threads (4 waves) = 1 M-tile x 4 N-tiles. grid.y covers Ncol/64.
// ---------------------------------------------------------------------------
__global__ void wmma_gemm_nt(const float* __restrict__ A, int lda,
                             long long aViewStride, int aViewWidth,
                             const float* __restrict__ B, int ldb,
                             float* __restrict__ C, int ldc,
                             int M, int Ncol, int K,
                             const float* __restrict__ bias, int fuse_elu)
{
    __shared__ float sA[16 * (TILE_K_MAX + 1)];
    const int KP = K + 1;
    const int m0 = blockIdx.x * 16;
    const int wave = threadIdx.x >> 5;
    const int lane = threadIdx.x & 31;
    const int n0 = (blockIdx.y * 4 + wave) * 16;

    // cooperative, coalesced load of the A tile [16 x K] into LDS
    for (int idx = threadIdx.x; idx < 16 * K; idx += blockDim.x) {
        int r = idx / K;
        int k = idx - r * K;
        int row = m0 + r;
        row = row < M ? row : M - 1;
        sA[r * KP + k] =
            A[(size_t)(k / aViewWidth) * (size_t)aViewStride +
              (size_t)row * lda + (k % aViewWidth)];
    }
    __syncthreads();
    if (n0 >= Ncol) return;

    v8f acc = {0.f, 0.f, 0.f, 0.f, 0.f, 0.f, 0.f, 0.f};
    const int rsel = lane & 15;           // A: M-in-tile / B: N-in-tile
    const int koff = (lane >> 4) << 1;    // K sub-offset {0,2}
    const float* Bcol = B + (size_t)(n0 + rsel) * ldb;

    for (int k0 = 0; k0 < K; k0 += 4) {
        v2f a, b;
        a.x = sA[rsel * KP + k0 + koff];
        a.y = sA[rsel * KP + k0 + koff + 1];
        b.x = Bcol[k0 + koff];
        b.y = Bcol[k0 + koff + 1];
        acc = __builtin_amdgcn_wmma_f32_16x16x4_f32(
            false, a, false, b, (short)0, acc, false, false);
    }

    // C/D layout: lane&15 = N, VGPR i -> M = i + 8*(lane>>4)
    const int colg = n0 + rsel;
    const float bs = bias ? bias[colg] : 0.f;
    const int mhi = (lane >> 4) * 8;
#pragma unroll
    for (int i = 0; i < 8; ++i) {
        int row = m0 + mhi + i;
        if (row < M) {
            float vout = acc[i] + bs;
            if (fuse_elu) vout = eluf(vout);
            C[(size_t)row * ldc + colg] = vout;
        }
    }
}

// ---------------------------------------------------------------------------
// Degree counting + dinv
// ---------------------------------------------------------------------------
__global__ void deg_count(const int* __restrict__ ei, float* __restrict__ deg,
                          int N, int E)
{
    long long t = (long long)blockIdx.x * blockDim.x + threadIdx.x;
    if (t >= 3LL * E) return;
    int v = (int)(t / E);
    int e = (int)(t - (long long)v * E);
    int col = ei[(size_t)v * 2 * E + E + e];
    atomicAdd(&deg[(size_t)v * N + col], 1.0f);
}

__global__ void dinv_k(float* __restrict__ deg, long long n)
{
    long long i = (long long)blockIdx.x * blockDim.x + threadIdx.x;
    if (i < n) deg[i] = rsqrtf(deg[i] + 1.0f);   // +1 = self loop; always > 0
}

// ---------------------------------------------------------------------------
// out[v][n][:] = b[v][:] + dinv[v][n]^2 * xlin[v][n][:]   (self loop + bias)
// ---------------------------------------------------------------------------
__global__ void self_bias_init(const float* __restrict__ xlin,
                               const float* __restrict__ dinv,
                               const float* __restrict__ bias,
                               float* __restrict__ out, int N, long long NH)
{
    long long idx = (long long)blockIdx.x * blockDim.x + threadIdx.x;
    if (idx >= 3 * NH) return;
    int v = (int)(idx / NH);
    long long r = idx - (long long)v * NH;
    int n = (int)(r >> 7);
    int h = (int)(r & 127);
    float d = dinv[(size_t)v * N + n];
    out[idx] = bias[v * 128 + h] + d * d * xlin[idx];
}

// ---------------------------------------------------------------------------
// Edge scatter-add: one wave32 per edge, 128 floats = 4 per lane.
// ---------------------------------------------------------------------------
__global__ void agg_edges(const float* __restrict__ xlin,
                          const int* __restrict__ ei,
                          const float* __restrict__ dinv,
                          float* __restrict__ out, int N, int E)
{
    long long w = (long long)blockIdx.x * (blockDim.x >> 5) + (threadIdx.x >> 5);
    if (w >= 3LL * E) return;
    int lane = threadIdx.x & 31;
    int v = (int)(w / E);
    int e = (int)(w - (long long)v * E);
    const int* eiv = ei + (size_t)v * 2 * E;
    int row = eiv[e];
    int col = eiv[E + e];
    float nrm = dinv[(size_t)v * N + row] * dinv[(size_t)v * N + col];
    const float* src = xlin + ((size_t)v * N + row) * 128;
    float* dst = out + ((size_t)v * N + col) * 128;
#pragma unroll
    for (int i = 0; i < 4; ++i) {
        int h = lane + 32 * i;
        atomicAdd(&dst[h], nrm * src[h]);
    }
}

// ---------------------------------------------------------------------------
// Attention over weight tensors: T = Ws@A, M[v][w] = <T[v], Ws[w]> + K*bA,
// row softmax of the tiny VxV matrix.
// ---------------------------------------------------------------------------
__global__ void att_T(const float* __restrict__ Ws, const float* __restrict__ Amat,
                      float* __restrict__ T, int Ddim)
{
    int vk = blockIdx.x;            // v*K + k
    int b = threadIdx.x;
    const float* wrow = Ws + (size_t)vk * Ddim;
    float s = 0.f;
    for (int a = 0; a < Ddim; ++a) s += wrow[a] * Amat[a * Ddim + b];
    T[(size_t)vk * Ddim + b] = s;
}

__global__ void att_M(const float* __restrict__ T, const float* __restrict__ Ws,
                      const float* __restrict__ bA, float* __restrict__ Mout,
                      int KD, int Kdim)
{
    int v = blockIdx.x / 3, w = blockIdx.x % 3;
    const float* tv = T + (size_t)v * KD;
    const float* ww = Ws + (size_t)w * KD;
    float s = 0.f;
    for (int i = threadIdx.x; i < KD; i += blockDim.x) s += tv[i] * ww[i];
    __shared__ float red[256];
    red[threadIdx.x] = s;
    __syncthreads();
    for (int off = 128; off > 0; off >>= 1) {
        if ((int)threadIdx.x < off) red[threadIdx.x] += red[threadIdx.x + off];
        __syncthreads();
    }
    if (threadIdx.x == 0) Mout[v * 3 + w] = red[0] + (float)Kdim * bA[0];
}

__global__ void att_softmax(const float* __restrict__ Mmat, float* __restrict__ att)
{
    int v = threadIdx.x;
    if (v >= 3) return;
    float mx = fmaxf(Mmat[v * 3], fmaxf(Mmat[v * 3 + 1], Mmat[v * 3 + 2]));
    float e0 = __expf(Mmat[v * 3 + 0] - mx);
    float e1 = __expf(Mmat[v * 3 + 1] - mx);
    float e2 = __expf(Mmat[v * 3 + 2] - mx);
    float s = e0 + e1 + e2;
    att[v * 3 + 0] = e0 / s;
    att[v * 3 + 1] = e1 / s;
    att[v * 3 + 2] = e2 / s;
}

// ---------------------------------------------------------------------------
// xr[v] = (1-alpha) + f(x[v]) + elu(alpha * sum_w att[v][w]*xlin[w]), in place.
// f = elu for layer 1, identity for layer 2. One thread per (n,h), all views.
// ---------------------------------------------------------------------------
__global__ void cross_rep_k(const float* __restrict__ xlin, float* __restrict__ xio,
                            const float* __restrict__ att, long long NH,
                            float alpha, int elu_x)
{
    long long i = (long long)blockIdx.x * blockDim.x + threadIdx.x;
    if (i >= NH) return;
    float l0 = xlin[i], l1 = xlin[NH + i], l2 = xlin[2 * NH + i];
    float a[9];
#pragma unroll
    for (int j = 0; j < 9; ++j) a[j] = att[j];
#pragma unroll
    for (int v = 0; v < 3; ++v) {
        float cr = alpha * (a[v * 3] * l0 + a[v * 3 + 1] * l1 + a[v * 3 + 2] * l2);
        cr = eluf(cr);
        float xi = xio[(long long)v * NH + i];
        if (elu_x) xi = eluf(xi);
        xio[(long long)v * NH + i] = (1.0f - alpha) + xi + cr;
    }
}

// ---------------------------------------------------------------------------
// Edge scoring: one wave per (view, edge, pos/neg); shuffle-reduced dot.
// ---------------------------------------------------------------------------
__global__ void score_k(const float* __restrict__ xr2, const int* __restrict__ epos,
                        const int* __restrict__ eneg, float* __restrict__ out,
                        int N, int Eev)
{
    long long w = (long long)blockIdx.x * (blockDim.x >> 5) + (threadIdx.x >> 5);
    long long total = 3LL * 2 * Eev;
    if (w >= total) return;
    int lane = threadIdx.x & 31;
    int v = (int)(w / (2 * Eev));
    int r = (int)(w - (long long)v * 2 * Eev);
    const int* ep = (r < Eev) ? epos : eneg;
    int e = (r < Eev) ? r : r - Eev;
    int na = ep[((size_t)v * Eev + e) * 2 + 0];
    int nb = ep[((size_t)v * Eev + e) * 2 + 1];
    const float* xa = xr2 + ((size_t)v * N + na) * 128;
    const float* xb = xr2 + ((size_t)v * N + nb) * 128;
    float s = 0.f;
#pragma unroll
    for (int i = 0; i < 4; ++i) {
        int h = lane + 32 * i;
        s += xa[h] * xb[h];
    }
#pragma unroll
    for (int off = 16; off > 0; off >>= 1) s += __shfl_xor(s, off, 32);
    if (lane == 0) out[(size_t)v * 2 * Eev + r] = s;
}

// ---------------------------------------------------------------------------
extern "C" void kernel_launch(void* const* d_in, const int* in_sizes, int n_in,
                              void* d_out, int out_size, void* d_ws, size_t ws_size,
                              hipStream_t stream)
{
    const float* x    = (const float*)d_in[0];
    const float* W1   = (const float*)d_in[1];
    const float* b1   = (const float*)d_in[2];
    const float* W2   = (const float*)d_in[3];
    const float* b2   = (const float*)d_in[4];
    const float* A1   = (const float*)d_in[5];
    const float* bA1  = (const float*)d_in[6];
    const float* A2   = (const float*)d_in[7];
    const float* bA2  = (const float*)d_in[8];
    const float* Wc1  = (const float*)d_in[9];
    const float* bc1  = (const float*)d_in[10];
    // d_in[11] (Wc2), d_in[12] (bc2) are unused by the reference computation
    const int* edge_index = (const int*)d_in[13];
    const int* edges      = (const int*)d_in[14];
    const int* edges_neg  = (const int*)d_in[15];

    const int H = 128, D = 128;
    const int N   = in_sizes[0] / D;
    const int E   = in_sizes[13] / 6;   // [3, 2, E]
    const int Eev = in_sizes[14] / 6;   // [3, Eev, 2]
    const long long NH = (long long)N * H;

    float* ws   = (float*)d_ws;
    float* bufA = ws;                          // 3*NH : xlin1 -> xlin2
    float* bufB = bufA + 3 * NH;               // 3*NH : x1/xr1 -> x2/xr2
    float* dinv = bufB + 3 * NH;               // 3*N
    float* xcb  = dinv + 3LL * N;              // NH
    float* Tbuf = xcb + NH;                    // 3*H*D
    float* Mmat = Tbuf + 3LL * H * D;          // 9
    float* attb = Mmat + 16;                   // 9

    const float alpha = 0.5f;

    // ---- degrees (shared by both layers) ----
    hipMemsetAsync(dinv, 0, (size_t)3 * N * sizeof(float), stream);
    {
        long long t = 3LL * E;
        deg_count<<<dim3((unsigned)((t + 255) / 256)), dim3(256), 0, stream>>>(
            edge_index, dinv, N, E);
        dinv_k<<<dim3((unsigned)((3LL * N + 255) / 256)), dim3(256), 0, stream>>>(
            dinv, 3LL * N);
    }

    const dim3 gemmBlk(128);
    const unsigned mt = (unsigned)((N + 15) / 16);

    // ---- layer 1: xlin1[v] = x @ W1[v]^T ----
    for (int v = 0; v < 3; ++v) {
        wmma_gemm_nt<<<dim3(mt, 2), gemmBlk, 0, stream>>>(
            x, D, 0LL, D, W1 + (size_t)v * H * D, D,
            bufA + (size_t)v * NH, H, N, H, D, nullptr, 0);
    }
    self_bias_init<<<dim3((unsigned)((3 * NH + 255) / 256)), dim3(256), 0, stream>>>(
        bufA, dinv, b1, bufB, N, NH);
    agg_edges<<<dim3((unsigned)((3LL * E + 7) / 8)), dim3(256), 0, stream>>>(
        bufA, edge_index, dinv, bufB, N, E);

    att_T<<<dim3(3 * H), dim3(D), 0, stream>>>(W1, A1, Tbuf, D);
    att_M<<<dim3(9), dim3(256), 0, stream>>>(Tbuf, W1, bA1, Mmat, H * D, H);
    att_softmax<<<dim3(1), dim3(32), 0, stream>>>(Mmat, attb);

    cross_rep_k<<<dim3((unsigned)((NH + 255) / 256)), dim3(256), 0, stream>>>(
        bufA, bufB, attb, NH, alpha, /*elu_x=*/1);

    // ---- combine: xc = elu(emb @ Wc1^T + bc1), emb[n, v*H+h] = xr1[v][n][h]
    wmma_gemm_nt<<<dim3(mt, 2), gemmBlk, 0, stream>>>(
        bufB, H, NH, H, Wc1, 3 * H, xcb, H, N, H, 3 * H, bc1, 1);

    // ---- layer 2: xlin2[v] = xc @ W2[v]^T ----
    for (int v = 0; v < 3; ++v) {
        wmma_gemm_nt<<<dim3(mt, 2), gemmBlk, 0, stream>>>(
            xcb, H, 0LL, H, W2 + (size_t)v * H * H, H,
            bufA + (size_t)v * NH, H, N, H, H, nullptr, 0);
    }
    self_bias_init<<<dim3((unsigned)((3 * NH + 255) / 256)), dim3(256), 0, stream>>>(
        bufA, dinv, b2, bufB, N, NH);
    agg_edges<<<dim3((unsigned)((3LL * E + 7) / 8)), dim3(256), 0, stream>>>(
        bufA, edge_index, dinv, bufB, N, E);

    att_T<<<dim3(3 * H), dim3(H), 0, stream>>>(W2, A2, Tbuf, H);
    att_M<<<dim3(9), dim3(256), 0, stream>>>(Tbuf, W2, bA2, Mmat, H * H, H);
    att_softmax<<<dim3(1), dim3(32), 0, stream>>>(Mmat, attb);

    cross_rep_k<<<dim3((unsigned)((NH + 255) / 256)), dim3(256), 0, stream>>>(
        bufA, bufB, attb, NH, alpha, /*elu_x=*/0);

    // ---- scoring ----
    score_k<<<dim3((unsigned)((6LL * Eev + 7) / 8)), dim3(256), 0, stream>>>(
        bufB, edges, edges_neg, (float*)d_out, N, Eev);
}